// QuantumSelfAttention_81535659147356
// MI455X (gfx1250) — compile-verified
//
#include <hip/hip_runtime.h>
#include <stdint.h>

// ---------------------------------------------------------------------------
// QuantumSelfAttention for MI455X (gfx1250, wave32)
//   v    = x @ Wv^T + bv                        (fp32 WMMA 16x16x4)
//   attn = softmax(threefry2x32-uniform * 0.125) fused with  attn @ v
//          (attn written non-temporally to d_out, consumed from LDS -> WMMA)
//   out  = (attn @ v) @ Wo^T + bo               (fp32 WMMA 16x16x4)
// Wq/bq/Wk/bk are dead in the reference and are skipped entirely.
// ---------------------------------------------------------------------------

typedef __attribute__((ext_vector_type(2))) float v2f;
typedef __attribute__((ext_vector_type(8))) float v8f;

#define B_BATCH 4
#define S_LEN   1024
#define TOTAL_D 1024
#define N_HEADS 16
#define HEAD_D  64
#define M_ROWS  (B_BATCH * S_LEN)                      // 4096
#define OUT0_ELEMS ((size_t)M_ROWS * TOTAL_D)          // 4,194,304
#define ATTN_ROWS  (B_BATCH * N_HEADS * S_LEN)         // 65,536
#define HALF_ROWS  (ATTN_ROWS / 2)                     // 32,768
#define HALF_ELEMS (1u << 25)                          // threefry counter split

// LDS score tile: [2 sets][16 rows][LDS_STRIDE]; stride 1028 => 16x4 A-frag
// (16 rows x 4 k-words = 64 words) hits all 64 banks exactly once, 8B-aligned.
#define LDS_STRIDE 1028
#define LDS_SCORES (2 * 16 * LDS_STRIDE)               // floats
#define LDS_SCRATCH (8 * 32 * 8)                       // 8 tiles x 32 lanes x 8
#define LDS_BYTES  ((LDS_SCORES + LDS_SCRATCH) * sizeof(float))

// ---------------------------------------------------------------------------
// threefry2x32, key = (0, 42)  == jax.random.key(42)
// ---------------------------------------------------------------------------
__device__ __forceinline__ uint32_t rotl32(uint32_t x, int r) {
    return (x << r) | (x >> (32 - r));
}

__device__ __forceinline__ void threefry2x32_k42(uint32_t x0, uint32_t x1,
                                                 uint32_t& o0, uint32_t& o1) {
    const uint32_t ks0 = 0u;
    const uint32_t ks1 = 42u;
    const uint32_t ks2 = 0u ^ 42u ^ 0x1BD11BDAu;
    x0 += ks0; x1 += ks1;
#define TF_ROUND(r) { x0 += x1; x1 = rotl32(x1, (r)); x1 ^= x0; }
    TF_ROUND(13) TF_ROUND(15) TF_ROUND(26) TF_ROUND(6)
    x0 += ks1; x1 += ks2 + 1u;
    TF_ROUND(17) TF_ROUND(29) TF_ROUND(16) TF_ROUND(24)
    x0 += ks2; x1 += ks0 + 2u;
    TF_ROUND(13) TF_ROUND(15) TF_ROUND(26) TF_ROUND(6)
    x0 += ks0; x1 += ks1 + 3u;
    TF_ROUND(17) TF_ROUND(29) TF_ROUND(16) TF_ROUND(24)
    x0 += ks1; x1 += ks2 + 4u;
    TF_ROUND(13) TF_ROUND(15) TF_ROUND(26) TF_ROUND(6)
    x0 += ks2; x1 += ks0 + 5u;
#undef TF_ROUND
    o0 = x0; o1 = x1;
}

__device__ __forceinline__ float bits_to_unit_float(uint32_t b) {
    // jax: bitcast((bits >> 9) | 0x3f800000) - 1.0  in [0, 1)
    return __uint_as_float((b >> 9) | 0x3f800000u) - 1.0f;
}

// ---------------------------------------------------------------------------
// GEMM: C[M,N] = A[M,K] @ W[N,K]^T + bias[N]   (fp32 WMMA 16x16x4)
// One wave -> one 32x32 C tile (2x2 accumulators).  (unchanged from round 1;
// codegen verified: pipelined global_load_b64 + v_wmma_f32_16x16x4_f32)
// ---------------------------------------------------------------------------
__global__ void gemm_at_bias_wmma(const float* __restrict__ A,
                                  const float* __restrict__ W,
                                  const float* __restrict__ bias,
                                  float* __restrict__ C,
                                  int M, int N, int K) {
    const int lane = threadIdx.x & 31;
    const int wid  = blockIdx.x * (blockDim.x >> 5) + (threadIdx.x >> 5);
    const int tilesN = N >> 5;
    const int tm = wid / tilesN;
    const int tn = wid - tm * tilesN;
    const int m0 = tm << 5;
    const int n0 = tn << 5;
    if (m0 >= M) return;

    const int lr = lane & 15;            // row within 16-frag
    const int kh = (lane >> 4) << 1;     // k sub-offset: 0 or 2

    const float* A0 = A + (size_t)(m0 + lr) * K + kh;
    const float* A1 = A0 + (size_t)16 * K;
    const float* B0 = W + (size_t)(n0 + lr) * K + kh;
    const float* B1 = B0 + (size_t)16 * K;

    v8f acc00 = {}, acc01 = {}, acc10 = {}, acc11 = {};

    for (int k = 0; k < K; k += 4) {
        v2f a0 = *(const v2f*)(A0 + k);
        v2f a1 = *(const v2f*)(A1 + k);
        v2f b0 = *(const v2f*)(B0 + k);
        v2f b1 = *(const v2f*)(B1 + k);
        acc00 = __builtin_amdgcn_wmma_f32_16x16x4_f32(false, a0, false, b0, (short)0, acc00, false, false);
        acc01 = __builtin_amdgcn_wmma_f32_16x16x4_f32(false, a0, false, b1, (short)0, acc01, false, false);
        acc10 = __builtin_amdgcn_wmma_f32_16x16x4_f32(false, a1, false, b0, (short)0, acc10, false, false);
        acc11 = __builtin_amdgcn_wmma_f32_16x16x4_f32(false, a1, false, b1, (short)0, acc11, false, false);
    }

    // C/D layout: VGPR t: lanes0-15 -> M = t, lanes16-31 -> M = 8+t; N = lane&15
    const int cn    = lane & 15;
    const int rbase = (lane >> 4) << 3;  // 0 or 8
    v8f accs[2][2] = {{acc00, acc01}, {acc10, acc11}};
    #pragma unroll
    for (int i = 0; i < 2; ++i) {
        #pragma unroll
        for (int j = 0; j < 2; ++j) {
            const int ncol = n0 + j * 16 + cn;
            const float bv = bias ? bias[ncol] : 0.0f;
            float* Cp = C + (size_t)(m0 + i * 16 + rbase) * N + ncol;
            #pragma unroll
            for (int t = 0; t < 8; ++t)
                Cp[(size_t)t * N] = accs[i][j][t] + bv;
        }
    }
}

// ---------------------------------------------------------------------------
// Fused: threefry scores -> softmax -> attn (NT store to d_out + LDS stage)
//        -> per-head  attn_tile[16x1024] @ v_head[1024x64]  via WMMA.
//
// Block = 512 threads (16 waves), grid = 2048:
//   wave w handles threefry row pair (R+w, R+w+32768); the block therefore
//   owns a 16x1024 attn tile of head R/1024 (set 0) and of head R/1024+32
//   (set 1).  Phase 2: 8 C-tiles (2 sets x 4 n-tiles of 16), 2 waves per
//   tile splitting K=1024; partials merged through LDS scratch.
// ---------------------------------------------------------------------------
__global__ void fused_score_softmax_attnv(float* __restrict__ attn,
                                          const float* __restrict__ vbuf,
                                          float* __restrict__ obuf) {
    extern __shared__ float lds[];
    const int lane = threadIdx.x & 31;
    const int w    = threadIdx.x >> 5;         // 0..15
    const int R    = blockIdx.x << 4;          // base row, set 0  (< 32768)
    const int r    = R + w;

    // ---------------- phase 1: threefry + dual softmax ----------------
    float e0[32], e1[32];
    float mx0 = -1e30f, mx1 = -1e30f;
    const uint32_t base = (uint32_t)r * 1024u + (uint32_t)lane;

    #pragma unroll
    for (int it = 0; it < 32; ++it) {
        const uint32_t i = base + (uint32_t)(it << 5);
        uint32_t o0, o1;
        threefry2x32_k42(i, i + HALF_ELEMS, o0, o1);
        const float s0 = bits_to_unit_float(o0) * 0.125f;   // * HEAD_DIM^-0.5
        const float s1 = bits_to_unit_float(o1) * 0.125f;
        e0[it] = s0; e1[it] = s1;
        mx0 = fmaxf(mx0, s0); mx1 = fmaxf(mx1, s1);
    }
    #pragma unroll
    for (int off = 16; off > 0; off >>= 1) {
        mx0 = fmaxf(mx0, __shfl_xor(mx0, off, 32));
        mx1 = fmaxf(mx1, __shfl_xor(mx1, off, 32));
    }
    float sum0 = 0.0f, sum1 = 0.0f;
    #pragma unroll
    for (int it = 0; it < 32; ++it) {
        e0[it] = expf(e0[it] - mx0); sum0 += e0[it];
        e1[it] = expf(e1[it] - mx1); sum1 += e1[it];
    }
    #pragma unroll
    for (int off = 16; off > 0; off >>= 1) {
        sum0 += __shfl_xor(sum0, off, 32);
        sum1 += __shfl_xor(sum1, off, 32);
    }
    const float inv0 = 1.0f / sum0;
    const float inv1 = 1.0f / sum1;

    float* l0 = lds + (size_t)(0 * 16 + w) * LDS_STRIDE + lane;
    float* l1 = lds + (size_t)(1 * 16 + w) * LDS_STRIDE + lane;
    float* row0 = attn + (size_t)r * 1024 + lane;
    float* row1 = attn + ((size_t)r + HALF_ROWS) * 1024 + lane;
    #pragma unroll
    for (int it = 0; it < 32; ++it) {
        const float a0 = e0[it] * inv0;
        const float a1 = e1[it] * inv1;
        l0[it << 5] = a0;
        l1[it << 5] = a1;
        __builtin_nontemporal_store(a0, row0 + (it << 5));  // write-once stream
        __builtin_nontemporal_store(a1, row1 + (it << 5));
    }
    __syncthreads();

    // ---------------- phase 2: attn_tile @ v_head (WMMA) ----------------
    const int tile  = w & 7;          // 0..7
    const int khalf = w >> 3;         // 0 or 1 : K range [khalf*512, +512)
    const int set   = tile >> 2;      // 0 or 1
    const int ntile = tile & 3;       // 0..3 -> n0 = ntile*16
    const int head  = (R >> 10) + set * 32;   // set0: 0..31, set1: 32..63
    const int b     = head >> 4;
    const int h     = head & 15;
    const int q0    = R & 1023;

    const int lr = lane & 15;
    const int kh = (lane >> 4) << 1;  // 0 or 2

    const float* ldsA = lds + (size_t)(set * 16 + lr) * LDS_STRIDE
                            + (khalf << 9) + kh;
    // B[k][n] = vbuf[(b*1024 + k)*1024 + h*64 + n]
    const float* B0 = vbuf + ((size_t)b * S_LEN + (khalf << 9) + kh) * TOTAL_D
                           + h * HEAD_D + (ntile << 4) + lr;

    v8f acc = {};
    for (int k = 0; k < 512; k += 4) {
        v2f a = *(const v2f*)(ldsA + k);                 // ds_load_b64
        const float* bp = B0 + (size_t)k * TOTAL_D;
        v2f bf;
        bf.x = bp[0];
        bf.y = bp[TOTAL_D];
        acc = __builtin_amdgcn_wmma_f32_16x16x4_f32(false, a, false, bf,
                                                    (short)0, acc, false, false);
    }

    // merge K-halves through LDS scratch, then write obuf ([B,S,H*D] layout)
    float* scratch = lds + LDS_SCORES + (size_t)(tile * 32 + lane) * 8;
    if (khalf == 1) {
        #pragma unroll
        for (int t = 0; t < 8; ++t) scratch[t] = acc[t];
    }
    __syncthreads();
    if (khalf == 0) {
        const int cn    = lane & 15;
        const int rbase = (lane >> 4) << 3;   // 0 or 8
        #pragma unroll
        for (int t = 0; t < 8; ++t) {
            const float val = acc[t] + scratch[t];
            const int m = q0 + rbase + t;
            obuf[((size_t)b * S_LEN + m) * TOTAL_D + h * HEAD_D + (ntile << 4) + cn] = val;
        }
    }
}

// ---------------------------------------------------------------------------
// Launch: inputs are  x, Wq, bq, Wk, bk, Wv, bv, Wo, bo  (all fp32).
// d_out = [final: 4096*1024] ++ [attn: 4*16*1024*1024]
// d_ws  = [vbuf: 16MB][obuf: 16MB]
// ---------------------------------------------------------------------------
extern "C" void kernel_launch(void* const* d_in, const int* in_sizes, int n_in,
                              void* d_out, int out_size, void* d_ws, size_t ws_size,
                              hipStream_t stream) {
    const float* x  = (const float*)d_in[0];
    const float* Wv = (const float*)d_in[5];
    const float* bv = (const float*)d_in[6];
    const float* Wo = (const float*)d_in[7];
    const float* bo = (const float*)d_in[8];

    float* out_final = (float*)d_out;
    float* attn      = (float*)d_out + OUT0_ELEMS;
    float* vbuf      = (float*)d_ws;
    float* obuf      = vbuf + OUT0_ELEMS;

    // 1) v = x @ Wv^T + bv : (4096/32)*(1024/32) = 4096 wave-tiles, 4/block
    gemm_at_bias_wmma<<<1024, 128, 0, stream>>>(x, Wv, bv, vbuf,
                                                M_ROWS, TOTAL_D, TOTAL_D);

    // 2+3) fused threefry->softmax->attn (NT store) + attn@v (LDS->WMMA)
    fused_score_softmax_attnv<<<2048, 512, LDS_BYTES, stream>>>(attn, vbuf, obuf);

    // 4) final = obuf @ Wo^T + bo
    gemm_at_bias_wmma<<<1024, 128, 0, stream>>>(obuf, Wo, bo, out_final,
                                                M_ROWS, TOTAL_D, TOTAL_D);
}